// MMNet_seg_80917183857402
// MI455X (gfx1250) — compile-verified
//
#include <hip/hip_runtime.h>
#include <math.h>

// ---------------- problem constants ----------------
#define BB    2
#define NPTS  65536          // N
#define KNN   16             // K neighbors
#define NK    (NPTS*KNN)     // 1048576 = 2^20
#define EPSBN 1e-5f

typedef float v2f __attribute__((ext_vector_type(2)));
typedef float v8f __attribute__((ext_vector_type(8)));

#if __has_builtin(__builtin_amdgcn_wmma_f32_16x16x4_f32)
#define HAVE_WMMA_F32X4 1
#else
#define HAVE_WMMA_F32X4 0
#endif

// D(16x16) = A(16x4) * B(4x16) + C, f32, wave32.
// A layout (ISA 7.12.2): lane l: M=l&15, VGPR0 -> K = 2*(l>>4), VGPR1 -> K+1.
// B layout (mirror):     lane l: N=l&15, VGPR0 -> K = 2*(l>>4), VGPR1 -> K+1.
// C/D: VGPR r: lanes 0-15 -> M=r, lanes 16-31 -> M=r+8, N=l&15.
__device__ __forceinline__ v8f wmma_f32_16x16x4(v2f a, v2f b, v8f c, int lane) {
#if HAVE_WMMA_F32X4
  (void)lane;
  return __builtin_amdgcn_wmma_f32_16x16x4_f32(false, a, false, b, (short)0, c,
                                               false, false);
#else
  // cross-lane emulation fallback (not used: builtin is present on gfx1250)
  int col = lane & 15;
  v8f d = c;
  #pragma unroll
  for (int r = 0; r < 8; ++r) {
    int row = r + ((lane >> 4) << 3);
    float acc = d[r];
    #pragma unroll
    for (int k = 0; k < 4; ++k) {
      int aSrc = row + ((k >> 1) << 4);
      int bSrc = col + ((k >> 1) << 4);
      float av = __shfl((k & 1) ? a.y : a.x, aSrc, 32);
      float bv = __shfl((k & 1) ? b.y : b.x, bSrc, 32);
      acc = fmaf(av, bv, acc);
    }
    d[r] = acc;
  }
  return d;
#endif
}

// ============================================================
// WMMA GEMM: yout[b,co,n] = sum_ci w[co,ci] * X[b,ci,n]
// X = xin, optionally transformed per-channel: relu(s[ci]*x + t[ci]).
// Also emits per-wave partial (sum, sumsq) per output channel for BN stats.
// ============================================================
template <int CO, int CI, bool XFORM>
__global__ void __launch_bounds__(256)
gemm_wmma_kernel(const float* __restrict__ w, const float* __restrict__ xin,
                 const float* __restrict__ bnp, float* __restrict__ yout,
                 float* __restrict__ partials) {
  constexpr int MT = (CO + 15) / 16;
  constexpr int KT = (CI + 3) / 4;
  const int lane  = threadIdx.x & 31;
  const int gwave = (int)((blockIdx.x * blockDim.x + threadIdx.x) >> 5);
  const int nwaves = (int)((gridDim.x * blockDim.x) >> 5);
  const int mrow  = lane & 15;
  const int khalf = (lane >> 4) << 1;
  const int hofs  = (lane >> 4) << 3;   // 0 for lanes 0-15, 8 for lanes 16-31

  // A fragments (weights), zero-padded to 16 rows / 4k steps
  v2f afrag[MT][KT];
  #pragma unroll
  for (int mt = 0; mt < MT; ++mt)
    #pragma unroll
    for (int kt = 0; kt < KT; ++kt) {
      int M = mt * 16 + mrow;
      int k = kt * 4 + khalf;
      afrag[mt][kt].x = (M < CO && k < CI) ? w[M * CI + k] : 0.f;
      afrag[mt][kt].y = (M < CO && (k + 1) < CI) ? w[M * CI + k + 1] : 0.f;
    }

  float scx[KT], scy[KT], shx[KT], shy[KT];
  if constexpr (XFORM) {
    #pragma unroll
    for (int kt = 0; kt < KT; ++kt) {
      int k = kt * 4 + khalf;
      scx[kt] = (k < CI) ? bnp[k] : 0.f;
      shx[kt] = (k < CI) ? bnp[32 + k] : 0.f;
      scy[kt] = (k + 1 < CI) ? bnp[k + 1] : 0.f;
      shy[kt] = (k + 1 < CI) ? bnp[32 + k + 1] : 0.f;
    }
  }

  float ssum[MT * 8], ssq[MT * 8];
  #pragma unroll
  for (int i = 0; i < MT * 8; ++i) { ssum[i] = 0.f; ssq[i] = 0.f; }

  const int ntiles = (BB * NPTS) / 16;
  for (int tile = gwave; tile < ntiles; tile += nwaves) {
    int p0 = tile * 16;
    int b  = p0 / NPTS;                 // tiles never straddle batches
    int n  = (p0 - b * NPTS) + mrow;
    v8f c[MT] = {};
    #pragma unroll
    for (int kt = 0; kt < KT; ++kt) {
      int k = kt * 4 + khalf;
      float x0 = (k < CI) ? xin[((size_t)(b * CI + k)) * NPTS + n] : 0.f;
      float x1 = ((k + 1) < CI) ? xin[((size_t)(b * CI + k + 1)) * NPTS + n] : 0.f;
      if constexpr (XFORM) {
        x0 = fmaxf(fmaf(scx[kt], x0, shx[kt]), 0.f);
        x1 = fmaxf(fmaf(scy[kt], x1, shy[kt]), 0.f);
      }
      v2f bf; bf.x = x0; bf.y = x1;
      #pragma unroll
      for (int mt = 0; mt < MT; ++mt)
        c[mt] = wmma_f32_16x16x4(afrag[mt][kt], bf, c[mt], lane);
    }
    #pragma unroll
    for (int mt = 0; mt < MT; ++mt) {
      // store predicate is uniform across r for a given lane: one branch,
      // not eight per-element saveexec chains
      if (mt * 16 + hofs < CO) {
        float* yb = yout + ((size_t)(b * CO + mt * 16 + hofs)) * NPTS + n;
        #pragma unroll
        for (int r = 0; r < 8; ++r) yb[(size_t)r * NPTS] = c[mt][r];
      }
      #pragma unroll
      for (int r = 0; r < 8; ++r) {
        float v = c[mt][r];
        ssum[mt * 8 + r] += v;
        ssq[mt * 8 + r] = fmaf(v, v, ssq[mt * 8 + r]);
      }
    }
  }

  // deterministic tree reduction inside each 16-lane half
  #pragma unroll
  for (int i = 0; i < MT * 8; ++i) {
    #pragma unroll
    for (int m = 1; m <= 8; m <<= 1) {
      ssum[i] += __shfl_xor(ssum[i], m, 32);
      ssq[i]  += __shfl_xor(ssq[i],  m, 32);
    }
  }
  if ((lane & 15) == 0) {
    float* pr = partials + (size_t)gwave * (2 * MT * 16);
    #pragma unroll
    for (int mt = 0; mt < MT; ++mt)
      #pragma unroll
      for (int r = 0; r < 8; ++r) {
        int ch = mt * 16 + r + hofs;
        pr[ch] = ssum[mt * 8 + r];
        pr[MT * 16 + ch] = ssq[mt * 8 + r];
      }
  }
}

// ============================================================
// BN finalize: mean/var from per-wave partials (fixed order -> deterministic),
// writes scale (out[0..C)) and shift (out[32..32+C)).  Biases cancel in
// training-mode BN, so they are omitted everywhere.
// ============================================================
__global__ void bn_finalize_kernel(const float* __restrict__ partials, int nP,
                                   int R, int sumOff, int sqOff, int C,
                                   float invCount, const float* __restrict__ g,
                                   const float* __restrict__ be,
                                   float* __restrict__ out) {
  int c = threadIdx.x;
  if (c >= C) return;
  float s = 0.f, s2 = 0.f;
  for (int i = 0; i < nP; ++i) {
    s  += partials[(size_t)i * R + sumOff + c];
    s2 += partials[(size_t)i * R + sqOff  + c];
  }
  float mean = s * invCount;
  float var  = fmaf(-mean, mean, s2 * invCount);
  float sc   = g[c] * (1.f / sqrtf(var + EPSBN));
  out[c]      = sc;
  out[32 + c] = fmaf(-sc, mean, be[c]);
}

// f = relu(bn1(y1)) elementwise over [B,C,N]
__global__ void bn_relu_apply_kernel(const float* __restrict__ y,
                                     const float* __restrict__ bnp,
                                     float* __restrict__ out, int C, int total) {
  int stride = gridDim.x * blockDim.x;
  for (int i = blockIdx.x * blockDim.x + threadIdx.x; i < total; i += stride) {
    int c = (i >> 16) % C;  // NPTS == 2^16
    out[i] = fmaxf(fmaf(bnp[c], y[i], bnp[32 + c]), 0.f);
  }
}

// out = leaky_relu(bn8(y8)), slope 0.01, layout [B,32,N]
__global__ void final_out_kernel(const float* __restrict__ y8,
                                 const float* __restrict__ bnp,
                                 float* __restrict__ out, int total) {
  int stride = gridDim.x * blockDim.x;
  for (int i = blockIdx.x * blockDim.x + threadIdx.x; i < total; i += stride) {
    int c = (i >> 16) & 31;
    float v = fmaf(bnp[c], y8[i], bnp[32 + c]);
    out[i] = (v > 0.f) ? v : 0.01f * v;
  }
}

// ============================================================
// P2: feat_info -> y2_pre = w2 @ [nf - tf, tf]   (+ stats, 3 ch)
// neigh_feat[b,c,k,n] = f[b, n&7, idx[b, c*NK/8 + k*N/8 + (n>>3)]]  (faithful reshape)
// ============================================================
__global__ void __launch_bounds__(256)
p2_kernel(const float* __restrict__ f, const int* __restrict__ idx,
          const float* __restrict__ w2g, float* __restrict__ y2,
          float* __restrict__ partials) {
  __shared__ float w2[48];
  for (int i = threadIdx.x; i < 48; i += blockDim.x) w2[i] = w2g[i];
  __syncthreads();
  float sum[3] = {0.f, 0.f, 0.f}, sq[3] = {0.f, 0.f, 0.f};
  const int total = BB * NK;
  const int stride = gridDim.x * blockDim.x;
  for (int t = blockIdx.x * blockDim.x + threadIdx.x; t < total; t += stride) {
    int b = t >> 20, rem = t & (NK - 1);
    int k = rem >> 16, n = rem & (NPTS - 1);
    const float* fb = f + (size_t)b * 8 * NPTS;
    const int* ib = idx + (size_t)b * NK;
    int cc = n & 7, rbase = k * (NPTS / 8) + (n >> 3);
    float nf[8], tf[8], fi[16];
    #pragma unroll
    for (int c = 0; c < 8; ++c) {
      int p = ib[c * (NK / 8) + rbase];
      nf[c] = fb[(size_t)cc * NPTS + p];
      tf[c] = fb[(size_t)c * NPTS + n];
      fi[c] = nf[c] - tf[c];
      fi[8 + c] = tf[c];
    }
    #pragma unroll
    for (int c3 = 0; c3 < 3; ++c3) {
      float acc = 0.f;
      #pragma unroll
      for (int j = 0; j < 16; ++j) acc = fmaf(w2[c3 * 16 + j], fi[j], acc);
      y2[(((size_t)(b * 3 + c3)) * KNN + k) * NPTS + n] = acc;
      sum[c3] += acc; sq[c3] = fmaf(acc, acc, sq[c3]);
    }
  }
  #pragma unroll
  for (int c3 = 0; c3 < 3; ++c3)
    #pragma unroll
    for (int m = 1; m < 32; m <<= 1) {
      sum[c3] += __shfl_xor(sum[c3], m, 32);
      sq[c3]  += __shfl_xor(sq[c3],  m, 32);
    }
  if ((threadIdx.x & 31) == 0) {
    int gwave = (int)((blockIdx.x * blockDim.x + threadIdx.x) >> 5);
    float* pr = partials + (size_t)gwave * 6;
    pr[0] = sum[0]; pr[1] = sum[1]; pr[2] = sum[2];
    pr[3] = sq[0];  pr[4] = sq[1];  pr[5] = sq[2];
  }
}

// ============================================================
// P3: shifted_xyz = neigh_xyz + relu(bn2(y2)) -> d_out (output #2);
//     y3 = w3 @ xyz_info (stats only), y4 = w4 @ xyz_info (stored + stats)
// neigh_xyz[b,c,k,n] = xyz[b, idx[b, t/3], t%3],  t = c*NK + k*N + n
// ============================================================
__global__ void __launch_bounds__(256)
p3_kernel(const float* __restrict__ xyz, const int* __restrict__ idx,
          const float* __restrict__ y2, const float* __restrict__ bn2,
          const float* __restrict__ w3g, const float* __restrict__ w4g,
          float* __restrict__ y4, float* __restrict__ outsx,
          float* __restrict__ partials) {
  __shared__ float w3[72], w4[72], s2[3], t2[3];
  for (int i = threadIdx.x; i < 72; i += blockDim.x) { w3[i] = w3g[i]; w4[i] = w4g[i]; }
  if (threadIdx.x < 3) { s2[threadIdx.x] = bn2[threadIdx.x]; t2[threadIdx.x] = bn2[32 + threadIdx.x]; }
  __syncthreads();
  float a3s[8], a3q[8], a4s[8], a4q[8];
  #pragma unroll
  for (int c = 0; c < 8; ++c) { a3s[c]=0.f; a3q[c]=0.f; a4s[c]=0.f; a4q[c]=0.f; }
  const int total = BB * NK;
  const int stride = gridDim.x * blockDim.x;
  for (int t = blockIdx.x * blockDim.x + threadIdx.x; t < total; t += stride) {
    int b = t >> 20, rem = t & (NK - 1);
    int k = rem >> 16, n = rem & (NPTS - 1);
    const float* xb = xyz + (size_t)b * NPTS * 3;
    const int* ib = idx + (size_t)b * NK;
    float nx[3], tx[3], sx[3];
    #pragma unroll
    for (int c3 = 0; c3 < 3; ++c3) {
      unsigned tt = (unsigned)c3 * NK + ((unsigned)k << 16) + (unsigned)n;
      unsigned r = tt / 3u, cc = tt - r * 3u;
      nx[c3] = xb[(size_t)ib[r] * 3 + cc];
      tx[c3] = xb[(size_t)n * 3 + c3];
      float yv = y2[(((size_t)(b * 3 + c3)) * KNN + k) * NPTS + n];
      sx[c3] = nx[c3] + fmaxf(fmaf(s2[c3], yv, t2[c3]), 0.f);
      outsx[(((size_t)(b * 3 + c3)) * KNN + k) * NPTS + n] = sx[c3];
    }
    float xi[9] = {nx[0]-tx[0], nx[1]-tx[1], nx[2]-tx[2],
                   sx[0], sx[1], sx[2], tx[0], tx[1], tx[2]};
    #pragma unroll
    for (int c = 0; c < 8; ++c) {
      float y3v = 0.f, y4v = 0.f;
      #pragma unroll
      for (int j = 0; j < 9; ++j) {
        y3v = fmaf(w3[c * 9 + j], xi[j], y3v);
        y4v = fmaf(w4[c * 9 + j], xi[j], y4v);
      }
      y4[(((size_t)(b * 8 + c)) * KNN + k) * NPTS + n] = y4v;
      a3s[c] += y3v; a3q[c] = fmaf(y3v, y3v, a3q[c]);
      a4s[c] += y4v; a4q[c] = fmaf(y4v, y4v, a4q[c]);
    }
  }
  #pragma unroll
  for (int c = 0; c < 8; ++c)
    #pragma unroll
    for (int m = 1; m < 32; m <<= 1) {
      a3s[c] += __shfl_xor(a3s[c], m, 32); a3q[c] += __shfl_xor(a3q[c], m, 32);
      a4s[c] += __shfl_xor(a4s[c], m, 32); a4q[c] += __shfl_xor(a4q[c], m, 32);
    }
  if ((threadIdx.x & 31) == 0) {
    int gwave = (int)((blockIdx.x * blockDim.x + threadIdx.x) >> 5);
    float* pr = partials + (size_t)gwave * 32;
    #pragma unroll
    for (int c = 0; c < 8; ++c) {
      pr[c] = a3s[c]; pr[8 + c] = a3q[c]; pr[16 + c] = a4s[c]; pr[24 + c] = a4q[c];
    }
  }
}

// ============================================================
// P4: recompute xyz_info (shifted_xyz read back from d_out), y3 = w3@xi,
// shifted_feat = nf + relu(bn3(y3)); y5 = w5 @ [sf, nf-tf, tf]  (+ stats)
// ============================================================
__global__ void __launch_bounds__(256)
p4_kernel(const float* __restrict__ xyz, const int* __restrict__ idx,
          const float* __restrict__ f, const float* __restrict__ sxin,
          const float* __restrict__ bn3, const float* __restrict__ w3g,
          const float* __restrict__ w5g, float* __restrict__ y5,
          float* __restrict__ partials) {
  __shared__ float w3[72], w5[192], s3[8], t3[8];
  for (int i = threadIdx.x; i < 72; i += blockDim.x) w3[i] = w3g[i];
  for (int i = threadIdx.x; i < 192; i += blockDim.x) w5[i] = w5g[i];
  if (threadIdx.x < 8) { s3[threadIdx.x] = bn3[threadIdx.x]; t3[threadIdx.x] = bn3[32 + threadIdx.x]; }
  __syncthreads();
  float a5s[8], a5q[8];
  #pragma unroll
  for (int c = 0; c < 8; ++c) { a5s[c] = 0.f; a5q[c] = 0.f; }
  const int total = BB * NK;
  const int stride = gridDim.x * blockDim.x;
  for (int t = blockIdx.x * blockDim.x + threadIdx.x; t < total; t += stride) {
    int b = t >> 20, rem = t & (NK - 1);
    int k = rem >> 16, n = rem & (NPTS - 1);
    const float* xb = xyz + (size_t)b * NPTS * 3;
    const float* fb = f + (size_t)b * 8 * NPTS;
    const int* ib = idx + (size_t)b * NK;
    float xi[9];
    #pragma unroll
    for (int c3 = 0; c3 < 3; ++c3) {
      unsigned tt = (unsigned)c3 * NK + ((unsigned)k << 16) + (unsigned)n;
      unsigned r = tt / 3u, cc = tt - r * 3u;
      float nx = xb[(size_t)ib[r] * 3 + cc];
      float tx = xb[(size_t)n * 3 + c3];
      xi[c3] = nx - tx;
      xi[3 + c3] = sxin[(((size_t)(b * 3 + c3)) * KNN + k) * NPTS + n];
      xi[6 + c3] = tx;
    }
    int cc = n & 7, rbase = k * (NPTS / 8) + (n >> 3);
    float fi2[24];
    #pragma unroll
    for (int c = 0; c < 8; ++c) {
      float y3v = 0.f;
      #pragma unroll
      for (int j = 0; j < 9; ++j) y3v = fmaf(w3[c * 9 + j], xi[j], y3v);
      int p = ib[c * (NK / 8) + rbase];
      float nf = fb[(size_t)cc * NPTS + p];
      float tf = fb[(size_t)c * NPTS + n];
      fi2[c] = nf + fmaxf(fmaf(s3[c], y3v, t3[c]), 0.f);  // shifted_feat
      fi2[8 + c] = nf - tf;
      fi2[16 + c] = tf;
    }
    #pragma unroll
    for (int c = 0; c < 8; ++c) {
      float acc = 0.f;
      #pragma unroll
      for (int j = 0; j < 24; ++j) acc = fmaf(w5[c * 24 + j], fi2[j], acc);
      y5[(((size_t)(b * 8 + c)) * KNN + k) * NPTS + n] = acc;
      a5s[c] += acc; a5q[c] = fmaf(acc, acc, a5q[c]);
    }
  }
  #pragma unroll
  for (int c = 0; c < 8; ++c)
    #pragma unroll
    for (int m = 1; m < 32; m <<= 1) {
      a5s[c] += __shfl_xor(a5s[c], m, 32);
      a5q[c] += __shfl_xor(a5q[c], m, 32);
    }
  if ((threadIdx.x & 31) == 0) {
    int gwave = (int)((blockIdx.x * blockDim.x + threadIdx.x) >> 5);
    float* pr = partials + (size_t)gwave * 16;
    #pragma unroll
    for (int c = 0; c < 8; ++c) { pr[c] = a5s[c]; pr[8 + c] = a5q[c]; }
  }
}

// ============================================================
// P5: per point (b,n): overall = [relu(bn4(y4)), relu(bn5(y5))] over k,
// kw = softmax_k(w6 @ overall) (b6 cancels), ws/mx pooling,
// enc = [mx(16), ws(16)] stored as [B,32,N].  Two passes over k keep regs low.
// ============================================================
__global__ void __launch_bounds__(256)
p5_kernel(const float* __restrict__ y4, const float* __restrict__ y5,
          const float* __restrict__ bn4, const float* __restrict__ bn5,
          const float* __restrict__ w6g, float* __restrict__ enc) {
  __shared__ float w6[256], s4[8], t4[8], s5[8], t5[8];
  for (int i = threadIdx.x; i < 256; i += blockDim.x) w6[i] = w6g[i];
  if (threadIdx.x < 8) {
    s4[threadIdx.x] = bn4[threadIdx.x]; t4[threadIdx.x] = bn4[32 + threadIdx.x];
    s5[threadIdx.x] = bn5[threadIdx.x]; t5[threadIdx.x] = bn5[32 + threadIdx.x];
  }
  __syncthreads();
  const int total = BB * NPTS;
  const int stride = gridDim.x * blockDim.x;
  for (int i = blockIdx.x * blockDim.x + threadIdx.x; i < total; i += stride) {
    int b = i >> 16, n = i & (NPTS - 1);
    const float* p4b = y4 + (size_t)b * 8 * KNN * NPTS + n;
    const float* p5b = y5 + (size_t)b * 8 * KNN * NPTS + n;
    float lmax[16];
    #pragma unroll
    for (int c = 0; c < 16; ++c) lmax[c] = -INFINITY;
    #pragma unroll 1
    for (int k = 0; k < KNN; ++k) {
      float ov[16];
      #pragma unroll
      for (int c = 0; c < 8; ++c) {
        ov[c]     = fmaxf(fmaf(s4[c], p4b[((size_t)c * KNN + k) * NPTS], t4[c]), 0.f);
        ov[8 + c] = fmaxf(fmaf(s5[c], p5b[((size_t)c * KNN + k) * NPTS], t5[c]), 0.f);
      }
      #pragma unroll
      for (int c = 0; c < 16; ++c) {
        float lg = 0.f;
        #pragma unroll
        for (int j = 0; j < 16; ++j) lg = fmaf(w6[c * 16 + j], ov[j], lg);
        lmax[c] = fmaxf(lmax[c], lg);
      }
    }
    float esum[16], wsum[16], mx[16];
    #pragma unroll
    for (int c = 0; c < 16; ++c) { esum[c] = 0.f; wsum[c] = 0.f; mx[c] = -INFINITY; }
    #pragma unroll 1
    for (int k = 0; k < KNN; ++k) {
      float ov[16];
      #pragma unroll
      for (int c = 0; c < 8; ++c) {
        ov[c]     = fmaxf(fmaf(s4[c], p4b[((size_t)c * KNN + k) * NPTS], t4[c]), 0.f);
        ov[8 + c] = fmaxf(fmaf(s5[c], p5b[((size_t)c * KNN + k) * NPTS], t5[c]), 0.f);
      }
      #pragma unroll
      for (int c = 0; c < 16; ++c) {
        float lg = 0.f;
        #pragma unroll
        for (int j = 0; j < 16; ++j) lg = fmaf(w6[c * 16 + j], ov[j], lg);
        float e = expf(lg - lmax[c]);
        esum[c] += e;
        wsum[c] = fmaf(ov[c], e, wsum[c]);
        mx[c] = fmaxf(mx[c], ov[c]);
      }
    }
    float* eb = enc + (size_t)b * 32 * NPTS + n;
    #pragma unroll
    for (int c = 0; c < 16; ++c) {
      eb[(size_t)c * NPTS] = mx[c];
      eb[(size_t)(16 + c) * NPTS] = wsum[c] / esum[c];
    }
  }
}

// ============================================================
extern "C" void kernel_launch(void* const* d_in, const int* in_sizes, int n_in,
                              void* d_out, int out_size, void* d_ws,
                              size_t ws_size, hipStream_t stream) {
  (void)in_sizes; (void)n_in; (void)out_size; (void)ws_size;
  // inputs in setup_inputs() dict insertion order (params flattened in order)
  const float* feature = (const float*)d_in[0];   // [B,32,1,N]
  const float* xyz     = (const float*)d_in[1];   // [B,N,3]
  const int*   nidx    = (const int*)d_in[2];     // [B,N*K]
  const float* w1 = (const float*)d_in[3];
  const float* w2 = (const float*)d_in[5];
  const float* w3 = (const float*)d_in[7];
  const float* w4 = (const float*)d_in[9];
  const float* w5 = (const float*)d_in[11];
  const float* w6 = (const float*)d_in[13];
  const float* w7 = (const float*)d_in[15];
  const float* w8 = (const float*)d_in[17];
  const float* g1 = (const float*)d_in[19]; const float* be1 = (const float*)d_in[20];
  const float* g2 = (const float*)d_in[21]; const float* be2 = (const float*)d_in[22];
  const float* g3 = (const float*)d_in[23]; const float* be3 = (const float*)d_in[24];
  const float* g4 = (const float*)d_in[25]; const float* be4 = (const float*)d_in[26];
  const float* g5 = (const float*)d_in[27]; const float* be5 = (const float*)d_in[28];
  const float* g7 = (const float*)d_in[29]; const float* be7 = (const float*)d_in[30];
  const float* g8 = (const float*)d_in[31]; const float* be8 = (const float*)d_in[32];

  // workspace carve-up (floats)
  float* ws  = (float*)d_ws;
  float* f   = ws;                       // [B,8,N]        1,048,576
  float* y1  = f + 1048576;              // [B,8,N]        1,048,576
  float* y2  = y1 + 1048576;             // [B,3,K,N]      6,291,456 (reused below)
  float* y7  = y2;                       // [B,16,N]       2,097,152 (after y2 dead)
  float* y8  = y2 + 2097152;             // [B,32,N]       4,194,304
  float* y4  = y2 + 6291456;             // [B,8,K,N]     16,777,216
  float* y5  = y4 + 16777216;            // [B,8,K,N]     16,777,216
  float* enc = y5 + 16777216;            // [B,32,N]       4,194,304
  float* part = enc + 4194304;           // partials         524,288
  float* bnA  = part + 524288;           // 7 slots x 64 floats
  float* L1 = bnA, *L2 = bnA + 64, *L3 = bnA + 128, *L4 = bnA + 192,
       * L5 = bnA + 256, *L7 = bnA + 320, *L8 = bnA + 384;

  float* out1 = (float*)d_out;           // [B,32,1,N]
  float* out2 = out1 + 4194304;          // shifted_xyz [B,3,K,N]

  const float invBN  = 1.f / (float)(BB * NPTS);       // layers 1,7,8
  const float invBKN = 1.f / (float)(BB * KNN * NPTS); // layers 2..5
  const int GW = 1024;   // waves for gemm kernels (128 blocks x 8 waves)
  const int PW = 8192;   // waves for pointwise kernels (1024 blocks x 8 waves)

  // conv1 (WMMA) + bn1 stats
  gemm_wmma_kernel<8, 32, false><<<128, 256, 0, stream>>>(w1, feature, nullptr, y1, part);
  bn_finalize_kernel<<<1, 64, 0, stream>>>(part, GW, 32, 0, 16, 8, invBN, g1, be1, L1);
  bn_relu_apply_kernel<<<512, 256, 0, stream>>>(y1, L1, f, 8, 8 * BB * NPTS);

  // conv2 + bn2 stats
  p2_kernel<<<1024, 256, 0, stream>>>(f, nidx, w2, y2, part);
  bn_finalize_kernel<<<1, 64, 0, stream>>>(part, PW, 6, 0, 3, 3, invBKN, g2, be2, L2);

  // shifted_xyz (output #2) + conv3/conv4 stats, y4 stored
  p3_kernel<<<1024, 256, 0, stream>>>(xyz, nidx, y2, L2, w3, w4, y4, out2, part);
  bn_finalize_kernel<<<1, 64, 0, stream>>>(part, PW, 32, 0, 8, 8, invBKN, g3, be3, L3);
  bn_finalize_kernel<<<1, 64, 0, stream>>>(part, PW, 32, 16, 24, 8, invBKN, g4, be4, L4);

  // conv5 + bn5 stats (y3 recomputed, shifted_xyz read back from d_out)
  p4_kernel<<<1024, 256, 0, stream>>>(xyz, nidx, f, out2, L3, w3, w5, y5, part);
  bn_finalize_kernel<<<1, 64, 0, stream>>>(part, PW, 16, 0, 8, 8, invBKN, g5, be5, L5);

  // attention pooling -> enc [B,32,N]
  p5_kernel<<<1024, 256, 0, stream>>>(y4, y5, L4, L5, w6, enc);

  // conv7 (WMMA) + bn7 stats
  gemm_wmma_kernel<16, 32, false><<<128, 256, 0, stream>>>(w7, enc, nullptr, y7, part);
  bn_finalize_kernel<<<1, 64, 0, stream>>>(part, GW, 32, 0, 16, 16, invBN, g7, be7, L7);

  // conv8 (WMMA, relu(bn7) folded into B-operand) + bn8 stats
  gemm_wmma_kernel<32, 16, true><<<128, 256, 0, stream>>>(w8, y7, L7, y8, part);
  bn_finalize_kernel<<<1, 64, 0, stream>>>(part, GW, 64, 0, 32, 32, invBN, g8, be8, L8);

  // out = leaky_relu(bn8(y8))
  final_out_kernel<<<1024, 256, 0, stream>>>(y8, L8, out1, 32 * BB * NPTS);
}